// Encoder_54168127537340
// MI455X (gfx1250) — compile-verified
//
#include <hip/hip_runtime.h>
#include <math.h>

// ---- problem constants (from reference) ----
#define B_   32
#define L_   256
#define D_   16
#define H_   64
#define K_   5
#define T_   252          // L - K + 1
#define P_   126          // T / 2
#define S_   8064         // P * H
#define OUT2S (2*S_)      // 16128

// ---- d_out flat layout (tuple concat in return order) ----
#define OUT_MAIN  0                       // (32,1,16128)
#define OUT_ATTN  (B_*OUT2S)              // 516096, (32,1,15)
#define OUT_RECON (OUT_ATTN + B_*(D_-1))  // 516576, (32,16,256)

// ---- workspace layout (floats) ----
#define WS_FLAT   0                         // (32,16,8064)
#define WS_MAINB  (B_*D_*S_)                // 4128768: contiguous main (32,8064)
#define WS_MPART  (WS_MAINB + B_*S_)        // 4386816: 4 x (32,8064) K-partials
#define MPART_STRIDE (B_*S_)                // 258048
#define NKPART    4
#define KCHUNK    (S_/NKPART)               // 2016

typedef __attribute__((ext_vector_type(2))) float v2f;
typedef __attribute__((ext_vector_type(8))) float v8f;

#define WMMA_F32X4(a, b, c) \
  __builtin_amdgcn_wmma_f32_16x16x4_f32(false, (a), false, (b), (short)0, (c), false, false)

// ---------------------------------------------------------------------------
// Kernel 1: per (b,d): conv+ReLU into LDS, then recon + pooled/flat/main.
// ---------------------------------------------------------------------------
__global__ void k_conv(const float* __restrict__ x,
                       const float* __restrict__ cwg,
                       const float* __restrict__ cbg,
                       const float* __restrict__ dwg,
                       const float* __restrict__ dbg,
                       float* __restrict__ out,
                       float* __restrict__ ws) {
  __shared__ float xs[L_];
  __shared__ float cw[H_*K_];
  __shared__ float cb[H_];
  __shared__ float dw[H_*K_];
  __shared__ float ys[H_*T_];   // 64*252 floats = 64.5 KB (of 320 KB)

  const int b = blockIdx.x >> 4;
  const int d = blockIdx.x & 15;
  const int tid = threadIdx.x;  // 256 threads

  xs[tid] = x[(b*L_ + tid)*D_ + d];
  if (tid < H_*K_) { cw[tid] = cwg[d*H_*K_ + tid]; dw[tid] = dwg[d*H_*K_ + tid]; }
  if (tid < H_)    { cb[tid] = cbg[d*H_ + tid]; }
  __syncthreads();

  // y[h][t] = relu(sum_k x[t+k]*cw[h][k] + cb[h])
  for (int idx = tid; idx < H_*T_; idx += 256) {
    const int h = idx / T_;
    const int t = idx - h*T_;
    float a = cb[h];
    const float* wp = &cw[h*K_];
#pragma unroll
    for (int k = 0; k < K_; ++k) a = fmaf(xs[t + k], wp[k], a);
    ys[idx] = fmaxf(a, 0.0f);
  }
  __syncthreads();

  // recon[b,d,l] = db[d] + sum_{h,j in [0,5): 0<=l-j<T} y[h][l-j] * dw[h][j]
  {
    const int l = tid;  // 256 outputs
    float a = dbg[d];
#pragma unroll
    for (int j = 0; j < K_; ++j) {
      const int t = l - j;
      if (t >= 0 && t < T_) {
        for (int h = 0; h < H_; ++h)
          a = fmaf(ys[h*T_ + t], dw[h*K_ + j], a);
      }
    }
    out[OUT_RECON + (b*D_ + d)*L_ + l] = a;
  }

  // flat[b,d, h*P+p] = 0.5*(y[h][2p] + y[h][2p+1]); contiguous main for d==0
  float* flat  = ws + WS_FLAT  + (size_t)(b*D_ + d)*S_;
  float* mainb = ws + WS_MAINB + (size_t)b*S_;
  for (int s = tid; s < S_; s += 256) {
    const int h = s / P_;
    const int p = s - h*P_;
    const float v = 0.5f*(ys[h*T_ + 2*p] + ys[h*T_ + 2*p + 1]);
    flat[s] = v;
    if (d == 0) mainb[s] = v;
  }
}

// ---------------------------------------------------------------------------
// Kernel 2: m = main(32x8064) @ W(8064x8064) via V_WMMA_F32_16X16X4_F32.
// Wave32: each wave -> 32x16 output tile (2 accumulators sharing B).
// Software-pipelined K loop, pointer-increment addressing, uniform prefetch
// guard (scalar branch, no EXEC churn). K split x4 -> private partial buffers.
// ---------------------------------------------------------------------------
__global__ void k_gemm(const float* __restrict__ A,   // main, (32,8064) row-major
                       const float* __restrict__ W,   // (8064,8064) row-major
                       float* __restrict__ mpart) {   // 4 x (32,8064)
  const int lane  = threadIdx.x & 31;
  const int wave  = threadIdx.x >> 5;          // 0..3
  const int n0    = (blockIdx.x*4 + wave)*16;  // N tile base
  const int kp    = blockIdx.y;                // 0..3
  const int l15   = lane & 15;
  const int khalf = lane >> 4;                 // 0 or 1

  const int kbeg = kp*KCHUNK;
  const int kend = kbeg + KCHUNK;

  // ISA 16x4 f32 A layout: lanes 0-15 rows M, VGPR0=K,K+2 halves; VGPR1=K+1,K+3
  const float* Ap0 = A + (size_t)l15*S_ + kbeg + 2*khalf;        // M = 0..15
  const float* Ap1 = Ap0 + (size_t)16*S_;                        // M = 16..31
  const float* Wp  = W + (size_t)(kbeg + 2*khalf)*S_ + n0 + l15; // B stream

  v8f c0 = {};
  v8f c1 = {};

  // prologue: operands for k0 = kbeg
  v2f a0 = *(const v2f*)Ap0;
  v2f a1 = *(const v2f*)Ap1;
  v2f bb;
  bb.x = Wp[0];
  bb.y = Wp[S_];

  const int kpf = kend - 68;  // uniform guard: row k0+66 stays inside chunk

  // pipelined main loop: fetch (k0+4) operands, then WMMA on k0 operands
#pragma unroll 2
  for (int k0 = kbeg; k0 < kend - 4; k0 += 4) {
    Ap0 += 4;
    Ap1 += 4;
    Wp  += (size_t)4*S_;
    v2f na0 = *(const v2f*)Ap0;
    v2f na1 = *(const v2f*)Ap1;
    v2f nbb;
    nbb.x = Wp[0];
    nbb.y = Wp[S_];
    if (k0 < kpf)                                  // scalar-uniform condition
      __builtin_prefetch(Wp + (size_t)60*S_, 0, 1);

    c0 = WMMA_F32X4(a0, bb, c0);
    c1 = WMMA_F32X4(a1, bb, c1);

    a0 = na0;
    a1 = na1;
    bb = nbb;
  }
  // epilogue: last K-step
  c0 = WMMA_F32X4(a0, bb, c0);
  c1 = WMMA_F32X4(a1, bb, c1);

  // C/D layout: VGPR j -> (M=j, N=lane) lanes 0-15 ; (M=j+8, N=lane-16) 16-31
  float* mp = mpart + (size_t)kp*MPART_STRIDE;
#pragma unroll
  for (int j = 0; j < 8; ++j) {
    const int r = j + 8*khalf;
    mp[(size_t)r*S_        + n0 + l15] = c0[j];
    mp[(size_t)(16 + r)*S_ + n0 + l15] = c1[j];
  }
}

// ---------------------------------------------------------------------------
// Kernel 3: scores -> softmax(15) -> weighted; assemble out + attn.
// One block per batch, 256 threads.
// ---------------------------------------------------------------------------
__global__ void k_attn(const float* __restrict__ ws, float* __restrict__ out) {
  const int b   = blockIdx.x;
  const int tid = threadIdx.x;

  const float* flat  = ws + WS_FLAT  + (size_t)b*D_*S_;
  const float* mainb = ws + WS_MAINB + (size_t)b*S_;
  const float* mp    = ws + WS_MPART + (size_t)b*S_;

  float acc[D_-1];
#pragma unroll
  for (int d = 0; d < D_-1; ++d) acc[d] = 0.0f;

  for (int s = tid; s < S_; s += 256) {
    const float mv = mp[s] + mp[MPART_STRIDE + s]
                   + mp[2*MPART_STRIDE + s] + mp[3*MPART_STRIDE + s];
#pragma unroll
    for (int d = 0; d < D_-1; ++d)
      acc[d] = fmaf(mv, flat[(size_t)(d + 1)*S_ + s], acc[d]);
  }

  // wave32 reduction, then cross-wave via LDS
#pragma unroll
  for (int d = 0; d < D_-1; ++d)
    for (int off = 16; off > 0; off >>= 1)
      acc[d] += __shfl_xor(acc[d], off, 32);

  __shared__ float red[8*(D_-1)];
  __shared__ float attnS[D_-1];
  const int lane = tid & 31, wv = tid >> 5;
  if (lane == 0) {
#pragma unroll
    for (int d = 0; d < D_-1; ++d) red[wv*(D_-1) + d] = acc[d];
  }
  __syncthreads();

  if (tid == 0) {
    float sc[D_-1];
#pragma unroll
    for (int d = 0; d < D_-1; ++d) {
      float s = 0.0f;
      for (int w = 0; w < 8; ++w) s += red[w*(D_-1) + d];
      sc[d] = s;
    }
    float mx = sc[0];
#pragma unroll
    for (int d = 1; d < D_-1; ++d) mx = fmaxf(mx, sc[d]);
    float sm = 0.0f;
#pragma unroll
    for (int d = 0; d < D_-1; ++d) { sc[d] = expf(sc[d] - mx); sm += sc[d]; }
    const float inv = 1.0f / sm;
#pragma unroll
    for (int d = 0; d < D_-1; ++d) {
      attnS[d] = sc[d] * inv;
      out[OUT_ATTN + b*(D_-1) + d] = attnS[d];
    }
  }
  __syncthreads();

  for (int s = tid; s < S_; s += 256) {
    float wsum = 0.0f;
#pragma unroll
    for (int d = 0; d < D_-1; ++d)
      wsum = fmaf(attnS[d], flat[(size_t)(d + 1)*S_ + s], wsum);
    out[(size_t)b*OUT2S + S_ + s] = wsum;     // weighted half
    out[(size_t)b*OUT2S + s]      = mainb[s]; // main half
  }
}

// ---------------------------------------------------------------------------
extern "C" void kernel_launch(void* const* d_in, const int* in_sizes, int n_in,
                              void* d_out, int out_size, void* d_ws, size_t ws_size,
                              hipStream_t stream) {
  const float* x  = (const float*)d_in[0];  // (32,256,16)
  const float* cw = (const float*)d_in[1];  // (16,64,5)
  const float* cb = (const float*)d_in[2];  // (16,64)
  const float* dw = (const float*)d_in[3];  // (16,64,5)
  const float* db = (const float*)d_in[4];  // (16,)
  const float* W  = (const float*)d_in[5];  // (8064,8064)
  float* out = (float*)d_out;
  float* ws  = (float*)d_ws;

  k_conv<<<dim3(B_*D_), dim3(256), 0, stream>>>(x, cw, cb, dw, db, out, ws);
  k_gemm<<<dim3(S_/64, NKPART), dim3(128), 0, stream>>>(ws + WS_MAINB, W,
                                                        ws + WS_MPART);
  k_attn<<<dim3(B_), dim3(256), 0, stream>>>(ws, out);
}